// PHNODE_54107997995500
// MI455X (gfx1250) — compile-verified
//
#include <hip/hip_runtime.h>

typedef float v2f __attribute__((ext_vector_type(2)));
typedef float v4f __attribute__((ext_vector_type(4)));
typedef float v8f __attribute__((ext_vector_type(8)));

#define MT 4                 // 16-row M tiles per workgroup (64 rows)
#define NT 8                 // 16-col N tiles per wave (128 cols strip)
#define BM (MT * 16)         // rows per workgroup
#define HID 1024

__global__ __launch_bounds__(256, 1)
void phnode_mlp_kernel(const float* __restrict__ y,
                       const float* __restrict__ control,
                       const float* __restrict__ mscal,
                       const float* __restrict__ W1,
                       const float* __restrict__ b1,
                       const float* __restrict__ W2,
                       const float* __restrict__ b2,
                       const float* __restrict__ W3,
                       const float* __restrict__ b3,
                       const float* __restrict__ R,
                       const float* __restrict__ Bmat,
                       const float* __restrict__ scale,
                       float* __restrict__ out)
{
    __shared__ float dH_part[2][BM];   // layer-3 cross-wave reduction buffer

    const int tid  = threadIdx.x;
    const int wave = tid >> 5;
    const int lane = tid & 31;
    const int l16  = lane & 15;
    const int half = lane >> 4;        // 0: lanes 0-15, 1: lanes 16-31
    const int kk   = half * 2;         // K sub-offset inside a K=4 step
    const long row0  = (long)blockIdx.x * BM;
    const int  nwave = wave * (NT * 16);   // this wave's 128-wide hidden strip

    if (tid < 2 * BM) ((float*)dH_part)[tid] = 0.0f;
    __syncthreads();

    // per-lane y for each M tile (row m = row0 + t*16 + l16)
    float y0[MT], y1[MT];
#pragma unroll
    for (int t = 0; t < MT; ++t) {
        const float* yp = y + (row0 + t * 16 + l16) * 2;
        y0[t] = yp[0];
        y1[t] = yp[1];
    }

    v8f acc[MT][NT];
    {
        v8f z = {0.f, 0.f, 0.f, 0.f, 0.f, 0.f, 0.f, 0.f};
#pragma unroll
        for (int t = 0; t < MT; ++t)
#pragma unroll
            for (int j = 0; j < NT; ++j)
                acc[t][j] = z;
    }

    // per-lane base into W2 for this wave's strip:
    // element W2[(nwave + l16)*HID + kk]
    const float* w2base = W2 + ((long)(nwave + l16) * HID + kk);

    for (int k0 = 0; k0 < HID; k0 += 4) {
        // ---- layer 1 fused: build A fragments on the fly ----
        const int k = k0 + kk;                          // lane's K column
        v4f w1v = *(const v4f*)(W1 + k * 2);            // W1[k][0..1], W1[k+1][0..1]
        v2f bv  = *(const v2f*)(b1 + k);                // b1[k], b1[k+1]
        v2f a[MT];
#pragma unroll
        for (int t = 0; t < MT; ++t) {
            float h0 = fmaf(y1[t], w1v.y, fmaf(y0[t], w1v.x, bv.x));
            float h1 = fmaf(y1[t], w1v.w, fmaf(y0[t], w1v.z, bv.y));
            a[t].x = fmaxf(h0, 0.0f);
            a[t].y = fmaxf(h1, 0.0f);
        }

        // ---- B fragments: gather W2[n, k..k+1] per lane ----
        v2f b[NT];
#pragma unroll
        for (int j = 0; j < NT; ++j)
            b[j] = *(const v2f*)(w2base + (long)j * (16 * HID) + k0);

        // ---- 32 f32 WMMAs per K step ----
#pragma unroll
        for (int t = 0; t < MT; ++t)
#pragma unroll
            for (int j = 0; j < NT; ++j)
                acc[t][j] = __builtin_amdgcn_wmma_f32_16x16x4_f32(
                    false, a[t], false, b[j], (short)0, acc[t][j], false, false);
    }

    // ---- layer 2 epilogue: + b2, relu (in registers) ----
    float b2v[NT];
#pragma unroll
    for (int j = 0; j < NT; ++j) b2v[j] = b2[nwave + j * 16 + l16];
#pragma unroll
    for (int t = 0; t < MT; ++t)
#pragma unroll
        for (int j = 0; j < NT; ++j)
#pragma unroll
            for (int r = 0; r < 8; ++r)
                acc[t][j][r] = fmaxf(acc[t][j][r] + b2v[j], 0.0f);

    // ---- layer 3 partial contraction: s[m] += W3[p,n] * h2[m,n] ----
    float w3v[2][NT];
#pragma unroll
    for (int p = 0; p < 2; ++p)
#pragma unroll
        for (int j = 0; j < NT; ++j)
            w3v[p][j] = W3[p * HID + nwave + j * 16 + l16];

#pragma unroll
    for (int p = 0; p < 2; ++p)
#pragma unroll
        for (int t = 0; t < MT; ++t) {
            float s[8];
#pragma unroll
            for (int r = 0; r < 8; ++r) s[r] = 0.0f;
#pragma unroll
            for (int j = 0; j < NT; ++j)
#pragma unroll
                for (int r = 0; r < 8; ++r)
                    s[r] = fmaf(w3v[p][j], acc[t][j][r], s[r]);
            // reduce over the 16 lanes that span N within this half-wave
#pragma unroll
            for (int r = 0; r < 8; ++r) {
                float v = s[r];
#pragma unroll
                for (int off = 8; off >= 1; off >>= 1)
                    v += __shfl_xor(v, off, 16);
                if (l16 == 0)
                    atomicAdd(&dH_part[p][t * 16 + r + 8 * half], v);
            }
        }
    __syncthreads();

    // ---- final 2x2 system per row ----
    if (tid < BM) {
        const float ms = mscal[0];
        const float u  = control[0];
        float d0 = dH_part[0][tid] + b3[0];
        float d1 = dH_part[1][tid] + b3[1];
        // A2 = J - R with J = [[0,-ms],[ms,0]]
        float a00 = -R[0];
        float a01 = -ms - R[1];
        float a10 =  ms - R[2];
        float a11 = -R[3];
        float x0 = fmaf(a00, d0, fmaf(a01, d1, Bmat[0] * u)) * scale[0];
        float x1 = fmaf(a10, d0, fmaf(a11, d1, Bmat[1] * u)) * scale[1];
        v2f o; o.x = x0; o.y = x1;
        *(v2f*)(out + (row0 + tid) * 2) = o;
    }
}

extern "C" void kernel_launch(void* const* d_in, const int* in_sizes, int n_in,
                              void* d_out, int out_size, void* d_ws, size_t ws_size,
                              hipStream_t stream) {
    const float* y       = (const float*)d_in[0];
    const float* control = (const float*)d_in[1];
    const float* mscal   = (const float*)d_in[2];
    const float* W1      = (const float*)d_in[3];
    const float* b1      = (const float*)d_in[4];
    const float* W2      = (const float*)d_in[5];
    const float* b2      = (const float*)d_in[6];
    const float* W3      = (const float*)d_in[7];
    const float* b3      = (const float*)d_in[8];
    const float* R       = (const float*)d_in[9];
    const float* Bmat    = (const float*)d_in[10];
    const float* scale   = (const float*)d_in[11];
    float* out = (float*)d_out;

    const int batch  = in_sizes[0] / 2;     // y is [B, 2]
    const int blocks = batch / BM;          // 131072 / 64 = 2048

    phnode_mlp_kernel<<<blocks, 256, 0, stream>>>(
        y, control, mscal, W1, b1, W2, b2, W3, b3, R, Bmat, scale, out);
}